// SPDTangentSpace_9835475107853
// MI455X (gfx1250) — compile-verified
//
#include <hip/hip_runtime.h>
#include <hip/hip_bf16.h>
#include <math.h>

// ---------------------------------------------------------------------------
// Batched SPD matrix logarithm for MI455X (gfx1250), B=16384, N=64.
// logm(X) via inverse scaling-and-squaring:
//   M0 = X/3  (eigs in [1/3, 5/3])
//   4 nested Newton-Schulz matrix square roots, iteration counts {7,6,5,5}
//     (from scalar error recurrence e' = (3e^2+e^3)/4; converged < fp32 eps)
//   E = M4 - I ; log(M4) ~= E*Q(E), Q = sum_{j=0..4} (-1)^j E^j/(j+1) + E^5/6 - E^6/7 (Horner)
//   logm(X) = 16*E*Q(E) + ln(3)*I
// All heavy work = 64x64x64 fp32 GEMMs on V_WMMA_F32_16X16X4_F32, data in LDS.
// 71 GEMMs/matrix; Z-update skipped on each root's last iteration.
// ---------------------------------------------------------------------------

typedef float v2f __attribute__((ext_vector_type(2)));
typedef float v8f __attribute__((ext_vector_type(8)));

#define NMAT 64
#define STRIDE 68              // 64 + 4 pad: bank = (68*i + j) % 64 = (4i+j)%64
#define MSZ (NMAT * STRIDE)    // floats per LDS matrix buffer

// C = alpha * (A x B) + beta * I   (64x64x64, fp32 WMMA, 8 waves / 256 thr)
__device__ __forceinline__ void mm64(const float* __restrict__ A,
                                     const float* __restrict__ B,
                                     float* __restrict__ C,
                                     float alpha, float beta, int tid) {
  const int wave = tid >> 5;
  const int lane = tid & 31;
  const int half = lane >> 4;   // 0: lanes 0-15, 1: lanes 16-31
  const int l16  = lane & 15;
#pragma unroll
  for (int tt = 0; tt < 2; ++tt) {          // 16 C-tiles over 8 waves
    const int t  = 2 * wave + tt;           // tile pair shares tm -> A reuse
    const int tm = t >> 2;
    const int tn = t & 3;
    v8f acc = {0.f, 0.f, 0.f, 0.f, 0.f, 0.f, 0.f, 0.f};
    // A 16x4 tile operand: lane holds row M=l16, K = 4k + 2*half + {0,1}
    const float* Ab = A + (tm * 16 + l16) * STRIDE + half * 2;
    // B 4x16 tile operand: lane holds col N=l16, K rows mirrored
    const float* Bb = B + (half * 2) * STRIDE + tn * 16 + l16;
#pragma unroll
    for (int k = 0; k < 16; ++k) {          // K = 64 in steps of 4
      v2f av, bv;
      av.x = Ab[4 * k + 0];
      av.y = Ab[4 * k + 1];
      bv.x = Bb[(4 * k + 0) * STRIDE];
      bv.y = Bb[(4 * k + 1) * STRIDE];
      acc = __builtin_amdgcn_wmma_f32_16x16x4_f32(false, av, false, bv,
                                                  (short)0, acc, false, false);
    }
    // C/D layout: VGPR r -> M = r (lanes 0-15) / r+8 (lanes 16-31), N = l16
    const int col = tn * 16 + l16;
#pragma unroll
    for (int r = 0; r < 8; ++r) {
      const int row = tm * 16 + r + half * 8;
      float v = alpha * acc[r];
      if (row == col) v += beta;
      C[row * STRIDE + col] = v;
    }
  }
}

__device__ __forceinline__ void set_identity(float* __restrict__ M, int tid) {
#pragma unroll
  for (int n = 0; n < 16; ++n) {
    const int e = tid + 256 * n;
    const int i = e >> 6, j = e & 63;
    M[i * STRIDE + j] = (i == j) ? 1.f : 0.f;
  }
}

__global__ __launch_bounds__(256) void spd_logm_kernel(
    const float* __restrict__ X, float* __restrict__ out, int nmat) {
  extern __shared__ float smem[];          // 4 * MSZ floats
  const int b   = blockIdx.x;
  const int tid = threadIdx.x;
  if (b >= nmat) return;

  // buffer indices into smem (no pointer table: avoids addrspacecast init)
  int bufY = 0, bufZ = 1, bufT = 2, bufU = 3;
#define BP(k) (smem + (k) * MSZ)

  // ---- load X/3 into Y (float4 global loads), Z = I ----
  {
    const float4* src = (const float4*)(X + (size_t)b * (NMAT * NMAT));
    float* Y = BP(bufY);
#pragma unroll
    for (int n = 0; n < 4; ++n) {
      const int e4 = tid + 256 * n;        // 1024 float4 = 4096 floats
      const float4 v = src[e4];
      const int i = e4 >> 4;               // 16 float4 per row
      const int j = (e4 & 15) * 4;
      float* d = &Y[i * STRIDE + j];
      const float s = (1.f / 3.f);
      d[0] = v.x * s; d[1] = v.y * s; d[2] = v.z * s; d[3] = v.w * s;
    }
    set_identity(BP(bufZ), tid);
  }
  __syncthreads();

  // ---- 4 nested Newton-Schulz square roots ----
  const int iters[4] = {7, 6, 5, 5};
  for (int root = 0; root < 4; ++root) {
    if (root > 0) {
      set_identity(BP(bufZ), tid);
      __syncthreads();
    }
    const int ni = iters[root];
    for (int it = 0; it < ni; ++it) {
      // T = 1.5 I - 0.5 (Z * Y)
      mm64(BP(bufZ), BP(bufY), BP(bufT), -0.5f, 1.5f, tid);
      __syncthreads();
      // Ynew = Y * T
      mm64(BP(bufY), BP(bufT), BP(bufU), 1.f, 0.f, tid);
      __syncthreads();
      if (it + 1 < ni) {
        // Znew = T * Z  (old Y buffer is free now)
        mm64(BP(bufT), BP(bufZ), BP(bufY), 1.f, 0.f, tid);
        __syncthreads();
        // rotate: Y<-U, Z<-oldY, U<-oldZ, T unchanged
        const int ny = bufU, nz = bufY, nu = bufZ;
        bufY = ny; bufZ = nz; bufU = nu;
      } else {
        // Z not needed past this root: new Y is in U, just swap Y<->U
        const int tmp = bufY; bufY = bufU; bufU = tmp;
      }
    }
  }

  // ---- E = M4 - I (in place in Y) ----
  {
    float* Y = BP(bufY);
#pragma unroll
    for (int n = 0; n < 16; ++n) {
      const int e = tid + 256 * n;
      const int i = e >> 6, j = e & 63;
      if (i == j) Y[i * STRIDE + j] -= 1.f;
    }
  }
  __syncthreads();

  // ---- Horner for Q(E) = I - E/2 + E^2/3 - E^3/4 + E^4/5 + E^5/6 - E^6/7  (order 7) ----
  // R5 = (1/6) I - (1/7) E   (elementwise init in Z)
  {
    float* Y = BP(bufY);
    float* Z = BP(bufZ);
#pragma unroll
    for (int n = 0; n < 16; ++n) {
      const int e = tid + 256 * n;
      const int i = e >> 6, j = e & 63;
      float v = (-1.f / 7.f) * Y[i * STRIDE + j];
      if (i == j) v += (1.f / 6.f);
      Z[i * STRIDE + j] = v;
    }
  }
  __syncthreads();

  const float cj[5] = {1.f / 5.f, -1.f / 4.f, 1.f / 3.f, -1.f / 2.f, 1.f};
  for (int s = 0; s < 5; ++s) {
    // R = E * R + c_j I
    mm64(BP(bufY), BP(bufZ), BP(bufT), 1.f, cj[s], tid);
    __syncthreads();
    const int tmp = bufZ; bufZ = bufT; bufT = tmp;
  }

  // ---- final: U = 16 * (E * Q) + ln(3) * I ----
  mm64(BP(bufY), BP(bufZ), BP(bufU), 16.f, 1.0986122886681098f, tid);
  __syncthreads();

  // ---- upper-triangle gather (np.triu_indices order) ----
  // base(i) = i*(129-i)/2 ; invert with closed form + fixup
  {
    const float* U = BP(bufU);
    float* dst = out + (size_t)b * 2080;
    for (int t = tid; t < 2080; t += 256) {
      int i = (int)floorf((129.f - sqrtf(129.f * 129.f - 8.f * (float)t)) * 0.5f);
      if (i < 0) i = 0;
      while (((i + 1) * (129 - (i + 1))) / 2 <= t) ++i;
      while ((i * (129 - i)) / 2 > t) --i;
      const int j = t - (i * (129 - i)) / 2 + i;
      dst[t] = U[i * STRIDE + j];
    }
  }
#undef BP
}

extern "C" void kernel_launch(void* const* d_in, const int* in_sizes, int n_in,
                              void* d_out, int out_size, void* d_ws, size_t ws_size,
                              hipStream_t stream) {
  (void)n_in; (void)d_ws; (void)ws_size; (void)out_size;
  const float* x = (const float*)d_in[0];
  float* out = (float*)d_out;
  const int nmat = in_sizes[0] / (NMAT * NMAT);          // 16384
  const size_t lds_bytes = 4u * MSZ * sizeof(float);     // 69632 B
  spd_logm_kernel<<<nmat, 256, lds_bytes, stream>>>(x, out, nmat);
}